// PredictNet_14181982011419
// MI455X (gfx1250) — compile-verified
//
#include <hip/hip_runtime.h>
#include <stdint.h>

// Problem constants (match reference setup_inputs()).
#define N_NODES 100000
#define N_REL   4
#define N_EDGE  500000
#define DIM     128

#define EPB     256     // edges staged per block
#define TPB     256     // threads per block (8 wave32s)

// ---------------------------------------------------------------------------
// Scatter: agg[dst] += val * x[src]  for one relation's COO edge list.
// One wave32 per edge: 32 lanes x float4 == one 512B (128-dim f32) row.
// Edge metadata (src/dst/val) for the block's chunk is staged into LDS with
// CDNA5 async global->LDS loads, so all pointer loads are issued in bulk and
// the per-edge reads in the hot loop are LDS broadcasts.
// (Defined first in the module so the disasm snippet shows this hot kernel.)
// ---------------------------------------------------------------------------
__global__ __launch_bounds__(TPB) void rgcn_scatter(
    const float* __restrict__ x,
    const int*   __restrict__ esrc,
    const int*   __restrict__ edst,
    const float* __restrict__ eval_,
    float*       __restrict__ agg,
    int n_edges)
{
    __shared__ int   s_src[EPB];
    __shared__ int   s_dst[EPB];
    __shared__ float s_val[EPB];

    const int tid  = threadIdx.x;
    const int base = blockIdx.x * EPB;
    const int idx  = base + tid;

    if (idx < n_edges) {
        // Low 32 bits of a generic LDS pointer are the LDS byte address.
        unsigned a_src = (unsigned)(uintptr_t)&s_src[tid];
        unsigned a_dst = (unsigned)(uintptr_t)&s_dst[tid];
        unsigned a_val = (unsigned)(uintptr_t)&s_val[tid];
        uint64_t g_src = (uint64_t)(uintptr_t)(esrc  + idx);
        uint64_t g_dst = (uint64_t)(uintptr_t)(edst  + idx);
        uint64_t g_val = (uint64_t)(uintptr_t)(eval_ + idx);
        asm volatile("global_load_async_to_lds_b32 %0, %1, off"
                     :: "v"(a_src), "v"(g_src) : "memory");
        asm volatile("global_load_async_to_lds_b32 %0, %1, off"
                     :: "v"(a_dst), "v"(g_dst) : "memory");
        asm volatile("global_load_async_to_lds_b32 %0, %1, off"
                     :: "v"(a_val), "v"(g_val) : "memory");
    }
    asm volatile("s_wait_asynccnt 0" ::: "memory");
    __syncthreads();

    const int wave = tid >> 5;   // 8 waves per block
    const int lane = tid & 31;

    #pragma unroll 4
    for (int it = 0; it < EPB / 8; ++it) {
        const int el = it * 8 + wave;     // edge within the staged chunk
        const int e  = base + el;
        if (e < n_edges) {
            // Wave-uniform LDS reads -> single-cycle broadcast.
            const int   s = s_src[el];
            const int   d = s_dst[el];
            const float v = s_val[el];

            // One global_load_b128 per lane: 32 lanes cover the 512B row.
            // Table is 51.2MB -> resident in the 192MB L2, so this hits L2.
            const float4 xv =
                reinterpret_cast<const float4*>(x + (size_t)s * DIM)[lane];

            float* dp = agg + (size_t)d * DIM + (size_t)lane * 4;
            // Relaxed agent-scope f32 adds -> want no-return global_atomic_add_f32
            // resolved at the L2 atomic units (STOREcnt only).
            __hip_atomic_fetch_add(dp + 0, v * xv.x, __ATOMIC_RELAXED, __HIP_MEMORY_SCOPE_AGENT);
            __hip_atomic_fetch_add(dp + 1, v * xv.y, __ATOMIC_RELAXED, __HIP_MEMORY_SCOPE_AGENT);
            __hip_atomic_fetch_add(dp + 2, v * xv.z, __ATOMIC_RELAXED, __HIP_MEMORY_SCOPE_AGENT);
            __hip_atomic_fetch_add(dp + 3, v * xv.w, __ATOMIC_RELAXED, __HIP_MEMORY_SCOPE_AGENT);
        }
    }
}

// ---------------------------------------------------------------------------
// Zero a float4 buffer.
// ---------------------------------------------------------------------------
__global__ __launch_bounds__(TPB) void rgcn_zero_f4(float4* __restrict__ p, int n4) {
    int i = blockIdx.x * TPB + threadIdx.x;
    if (i < n4) {
        p[i] = make_float4(0.f, 0.f, 0.f, 0.f);
    }
}

// ---------------------------------------------------------------------------
// out (+)= leaky_relu(agg); agg = 0   (fused clear saves a zeroing pass)
// ---------------------------------------------------------------------------
__global__ __launch_bounds__(TPB) void rgcn_accum_relu_clear(
    float4* __restrict__ agg, float4* __restrict__ out, int n4, int first)
{
    int i = blockIdx.x * TPB + threadIdx.x;
    if (i < n4) {
        float4 a = agg[i];
        float4 y;
        y.x = a.x > 0.f ? a.x : 0.01f * a.x;
        y.y = a.y > 0.f ? a.y : 0.01f * a.y;
        y.z = a.z > 0.f ? a.z : 0.01f * a.z;
        y.w = a.w > 0.f ? a.w : 0.01f * a.w;
        if (!first) {
            float4 o = out[i];
            y.x += o.x; y.y += o.y; y.z += o.z; y.w += o.w;
        }
        out[i] = y;
        agg[i] = make_float4(0.f, 0.f, 0.f, 0.f);
    }
}

// ---------------------------------------------------------------------------
// Launch: two RGCN layers, 4 relations each, sequential on `stream`.
//   ws layout: [ agg : N*D floats ][ xmid : N*D floats ]   (~102.4 MB)
// ---------------------------------------------------------------------------
extern "C" void kernel_launch(void* const* d_in, const int* in_sizes, int n_in,
                              void* d_out, int out_size, void* d_ws, size_t ws_size,
                              hipStream_t stream) {
    const float* embs  = (const float*)d_in[0];
    const int*   esrc  = (const int*)  d_in[1];   // [N_REL, N_EDGE]
    const int*   edst  = (const int*)  d_in[2];   // [N_REL, N_EDGE]
    const float* eval_ = (const float*)d_in[3];   // [N_REL, N_EDGE]
    float* out  = (float*)d_out;                  // [N_NODES, DIM]

    float* agg  = (float*)d_ws;
    float* xmid = agg + (size_t)N_NODES * DIM;

    const int n4          = N_NODES * DIM / 4;            // 3,200,000
    const int dense_grid  = (n4 + TPB - 1) / TPB;          // 12,500
    const int scat_grid   = (N_EDGE + EPB - 1) / EPB;      // 1,954

    // agg must be zero at the start of every launch (ws is poisoned once and
    // we must not rely on state left behind by a previous call).
    rgcn_zero_f4<<<dense_grid, TPB, 0, stream>>>((float4*)agg, n4);

    for (int layer = 0; layer < 2; ++layer) {
        const float* xin  = (layer == 0) ? embs : xmid;
        float*       xout = (layer == 0) ? xmid : out;
        for (int r = 0; r < N_REL; ++r) {
            rgcn_scatter<<<scat_grid, TPB, 0, stream>>>(
                xin,
                esrc  + (size_t)r * N_EDGE,
                edst  + (size_t)r * N_EDGE,
                eval_ + (size_t)r * N_EDGE,
                agg, N_EDGE);
            rgcn_accum_relu_clear<<<dense_grid, TPB, 0, stream>>>(
                (float4*)agg, (float4*)xout, n4, (r == 0) ? 1 : 0);
        }
    }
}